// TaylorSeriesLinearAttn_42039139894035
// MI455X (gfx1250) — compile-verified
//
#include <hip/hip_runtime.h>
#include <hip/hip_bf16.h>

// ---------------------------------------------------------------------------
// Taylor-series causal linear attention on gfx1250 (CDNA5, wave32, WMMA).
// f32 end-to-end using V_WMMA_F32_16X16X4_F32 (exact-precision matrix path).
// Chunked formulation; feature dim D=288 split across 6 waves per head.
// ---------------------------------------------------------------------------

typedef __attribute__((ext_vector_type(2))) float v2f;
typedef __attribute__((ext_vector_type(8))) float v8f;

#define N_TOK   2048
#define DIM_X   512
#define HEADS   8
#define DH      16
#define DFEAT   273          // 1 + 16 + 256
#define DPAD    288          // padded to multiple of 16
#define LQ      289          // LDS row stride for qf/kf (289 % 64 = 33, odd)
#define NW      6            // waves per head-block (6 * 48 = 288)
#define KSLICE  48           // feature columns per wave
#define SCALE   0.25f        // 16^-0.5
#define SQRTH   0.70710678118654752f
#define EPSV    1e-5f

static __device__ __forceinline__ v8f wmma4(v2f a, v2f b, v8f c) {
    // D = A(16x4,f32) * B(4x16,f32) + C(16x16,f32)
    return __builtin_amdgcn_wmma_f32_16x16x4_f32(
        false, a, false, b, (short)0, c, false, false);
}

// ---------------------------------------------------------------------------
// Kernel 1: projections  q = (x@w_q)*SCALE , k,v from x@w_kv
// One block per 16-row tile of x; x panel staged once in LDS; 8 waves cover
// 8 of 24 column tiles (grid.y = 3).
// output layout: qs/ks/vs [HEADS][N_TOK][DH]
// ---------------------------------------------------------------------------
__global__ __launch_bounds__(256)
void proj_kernel(const float* __restrict__ x,
                 const float* __restrict__ w_q,
                 const float* __restrict__ w_kv,
                 float* __restrict__ qs,
                 float* __restrict__ ks,
                 float* __restrict__ vs) {
    __shared__ float xs[16 * 513];              // stride 513 % 64 == 1

    const int t    = threadIdx.x;
    const int wid  = t >> 5;
    const int l    = t & 31;
    const int lm   = l & 15;
    const int half = l >> 4;

    const int row0 = blockIdx.x * 16;
    const int ct   = blockIdx.y * 8 + wid;      // 0..23 column tiles

    // stage x panel [16 x 512] in LDS (coalesced)
    for (int idx = t; idx < 16 * DIM_X; idx += 256) {
        int r = idx >> 9, c = idx & 511;
        xs[r * 513 + c] = x[(size_t)(row0 + r) * DIM_X + c];
    }
    __syncthreads();

    const float* B;
    int ldb, col0, h;
    float scl;
    float* dst;
    if (ct < 8)       { B = w_q;  ldb = 128; col0 = ct * 16;              h = ct;      scl = SCALE; dst = qs; }
    else if (ct < 16) { B = w_kv; ldb = 256; col0 = (ct - 8) * 16;        h = ct - 8;  scl = 1.0f;  dst = ks; }
    else              { B = w_kv; ldb = 256; col0 = 128 + (ct - 16) * 16; h = ct - 16; scl = 1.0f;  dst = vs; }

    v8f acc = {};
    for (int kk = 0; kk < DIM_X; kk += 4) {
        if ((kk & 63) == 0 && kk + 64 < DIM_X)  // gfx1250 global_prefetch_b8
            __builtin_prefetch(&B[(size_t)(kk + 64 + 2 * half) * ldb + col0 + lm], 0, 1);
        v2f a, b;
        a.x = xs[lm * 513 + kk + 2 * half];
        a.y = xs[lm * 513 + kk + 2 * half + 1];
        const float* bp = B + (size_t)(kk + 2 * half) * ldb + col0 + lm;
        b.x = bp[0];
        b.y = bp[ldb];
        acc = wmma4(a, b, acc);
    }

    float* out = dst + (size_t)h * N_TOK * DH;
#pragma unroll
    for (int i = 0; i < 8; ++i) {
        int M = i + 8 * half;
        out[(size_t)(row0 + M) * DH + lm] = acc[i] * scl;
    }
}

// ---------------------------------------------------------------------------
// Kernel 2: chunked causal linear attention. One block (6 waves) per head,
// 128 sequential chunks of 16 positions. State S[DPAD][16], z[DPAD] in LDS.
// Wave w owns feature slice [w*48, w*48+48):
//   partial A     = qf_slice @ kf_slice^T        (sum-reduced across waves)
//   partial num   = qf_slice @ S_slice           (sum-reduced across waves)
//   state update  S_slice += kf_slice^T @ V      (no reduction needed)
// den[m] = qf[m].z + rowsum(masked A[m])  (exact identity with A).
// attn output layout: [N_TOK][HEADS*DH]
// ---------------------------------------------------------------------------
__global__ __launch_bounds__(NW * 32)
void attn_kernel(const float* __restrict__ qs,
                 const float* __restrict__ ks,
                 const float* __restrict__ vs,
                 float* __restrict__ attn) {
    __shared__ float qf[16 * LQ];
    __shared__ float kf[16 * LQ];
    __shared__ float S [DPAD * DH];
    __shared__ float z [DPAD];
    __shared__ float qt[16 * DH];
    __shared__ float kt[16 * DH];
    __shared__ float vt[16 * DH];
    __shared__ float Apart[NW * 256];
    __shared__ float Npart[NW * 256];
    __shared__ float dpart[NW * 16];
    __shared__ float Ast[256];
    __shared__ float dden[16];

    const int h     = blockIdx.x;
    const int t     = threadIdx.x;              // 0..191
    const int wid   = t >> 5;
    const int l     = t & 31;
    const int lm    = l & 15;
    const int half  = l >> 4;
    const int kbase = wid * KSLICE;             // this wave's feature slice

    const float* qh = qs + (size_t)h * N_TOK * DH;
    const float* kh = ks + (size_t)h * N_TOK * DH;
    const float* vh = vs + (size_t)h * N_TOK * DH;

    for (int i = t; i < DPAD * DH; i += NW * 32) S[i] = 0.0f;
    for (int i = t; i < DPAD;      i += NW * 32) z[i] = 0.0f;
    __syncthreads();

    for (int c = 0; c < N_TOK / 16; ++c) {
        const int pos0 = c * 16;

        // ---- load raw 16x16 q/k/v tiles ----
        for (int idx = t; idx < 256; idx += NW * 32) {
            int m = idx >> 4, d = idx & 15;
            qt[idx] = qh[(size_t)(pos0 + m) * DH + d];
            kt[idx] = kh[(size_t)(pos0 + m) * DH + d];
            vt[idx] = vh[(size_t)(pos0 + m) * DH + d];
        }
        __syncthreads();

        // ---- second-order Taylor feature expansion ----
        for (int idx = t; idx < 16 * DPAD; idx += NW * 32) {
            int m = idx / DPAD, j = idx % DPAD;
            float vq, vk;
            if (j == 0)          { vq = 1.0f; vk = 1.0f; }
            else if (j < 17)     { vq = qt[m * 16 + (j - 1)];
                                   vk = kt[m * 16 + (j - 1)]; }
            else if (j < DFEAT)  { int p = j - 17, i2 = p >> 4, j2 = p & 15;
                                   vq = qt[m * 16 + i2] * qt[m * 16 + j2] * SQRTH;
                                   vk = kt[m * 16 + i2] * kt[m * 16 + j2] * SQRTH; }
            else                 { vq = 0.0f; vk = 0.0f; }
            qf[m * LQ + j] = vq;
            kf[m * LQ + j] = vk;
        }
        __syncthreads();

        // ---- per-wave partial A = qf@kf^T and partial num = qf@S_prev ----
        v8f accA = {};
        v8f accN = {};
#pragma unroll
        for (int kk = 0; kk < KSLICE; kk += 4) {
            int k0 = kbase + kk;
            v2f a, bA, bS;
            a.x  = qf[lm * LQ + k0 + 2 * half];
            a.y  = qf[lm * LQ + k0 + 2 * half + 1];
            bA.x = kf[lm * LQ + k0 + 2 * half];       // (kf^T)[k][n] = kf[n][k]
            bA.y = kf[lm * LQ + k0 + 2 * half + 1];
            bS.x = S[(k0 + 2 * half) * DH + lm];
            bS.y = S[(k0 + 2 * half + 1) * DH + lm];
            accA = wmma4(a, bA, accA);
            accN = wmma4(a, bS, accN);
        }
#pragma unroll
        for (int i = 0; i < 8; ++i)
            Apart[wid * 256 + (i + 8 * half) * 16 + lm] = accA[i];
        if (wid != 0) {
#pragma unroll
            for (int i = 0; i < 8; ++i)
                Npart[wid * 256 + (i + 8 * half) * 16 + lm] = accN[i];
        }
        // partial qf.z over this wave's slice (reads z of previous chunks)
        if (l < 16) {
            float s = 0.0f;
#pragma unroll
            for (int jj = 0; jj < KSLICE; ++jj) {
                int j = kbase + jj;
                if (j < DFEAT) s += qf[l * LQ + j] * z[j];
            }
            dpart[wid * 16 + l] = s;
        }
        __syncthreads();

        // ---- reduce A across waves + causal mask ----
        for (int idx = t; idx < 256; idx += NW * 32) {
            int M = idx >> 4, N = idx & 15;
            float s = 0.0f;
#pragma unroll
            for (int w = 0; w < NW; ++w) s += Apart[w * 256 + idx];
            Ast[idx] = (N <= M) ? s : 0.0f;
        }
        __syncthreads();

        // ---- wave 0: intra-chunk num += A_masked @ V ; den ----
        if (wid == 0) {
#pragma unroll
            for (int k0 = 0; k0 < 16; k0 += 4) {
                v2f a, b;
                a.x = Ast[lm * 16 + k0 + 2 * half];
                a.y = Ast[lm * 16 + k0 + 2 * half + 1];
                b.x = vt[(k0 + 2 * half) * 16 + lm];
                b.y = vt[(k0 + 2 * half + 1) * 16 + lm];
                accN = wmma4(a, b, accN);
            }
#pragma unroll
            for (int i = 0; i < 8; ++i)
                Npart[(i + 8 * half) * 16 + lm] = accN[i];
            if (l < 16) {
                float qz = 0.0f;
#pragma unroll
                for (int w = 0; w < NW; ++w) qz += dpart[w * 16 + l];
                float rs = 0.0f;
#pragma unroll
                for (int n2 = 0; n2 < 16; ++n2) rs += Ast[l * 16 + n2];
                dden[l] = fmaxf(qz + rs, EPSV);
            }
        }

        // ---- state update: S_slice += kf_slice^T @ V  (3 tiles per wave) ----
#pragma unroll
        for (int jt3 = 0; jt3 < 3; ++jt3) {
            int jt = wid * 3 + jt3;             // 0..17
            v8f cS;
#pragma unroll
            for (int i = 0; i < 8; ++i)
                cS[i] = S[(jt * 16 + i + 8 * half) * DH + lm];
#pragma unroll
            for (int k0 = 0; k0 < 16; k0 += 4) {
                v2f a, b;
                // A[j][m] = kf[m][j]
                a.x = kf[(k0 + 2 * half) * LQ + jt * 16 + lm];
                a.y = kf[(k0 + 2 * half + 1) * LQ + jt * 16 + lm];
                b.x = vt[(k0 + 2 * half) * 16 + lm];
                b.y = vt[(k0 + 2 * half + 1) * 16 + lm];
                cS = wmma4(a, b, cS);
            }
#pragma unroll
            for (int i = 0; i < 8; ++i)
                S[(jt * 16 + i + 8 * half) * DH + lm] = cS[i];
        }

        // ---- z update over this wave's slice ----
        for (int jj = l; jj < KSLICE; jj += 32) {
            int j = kbase + jj;
            float s = 0.0f;
#pragma unroll
            for (int m = 0; m < 16; ++m) s += kf[m * LQ + j];
            z[j] += s;
        }
        __syncthreads();

        // ---- reduce num across waves, divide by den, store ----
        for (int idx = t; idx < 256; idx += NW * 32) {
            int M = idx >> 4, e = idx & 15;
            float num = 0.0f;
#pragma unroll
            for (int w = 0; w < NW; ++w) num += Npart[w * 256 + idx];
            attn[(size_t)(pos0 + M) * (HEADS * DH) + h * DH + e] = num / dden[M];
        }
        __syncthreads();
    }
}

// ---------------------------------------------------------------------------
// Kernel 3: output projection  out = attn[2048,128] @ w_out[128,512]
// One block per 16-row tile; attn panel staged in LDS; 8 waves cover 8 of
// 32 column tiles (grid.y = 4).
// ---------------------------------------------------------------------------
__global__ __launch_bounds__(256)
void outproj_kernel(const float* __restrict__ attn,
                    const float* __restrict__ w_out,
                    float* __restrict__ out) {
    __shared__ float as[16 * 129];              // stride 129 % 64 == 1

    const int t    = threadIdx.x;
    const int wid  = t >> 5;
    const int l    = t & 31;
    const int lm   = l & 15;
    const int half = l >> 4;

    const int row0 = blockIdx.x * 16;
    const int col0 = (blockIdx.y * 8 + wid) * 16;

    for (int idx = t; idx < 16 * 128; idx += 256) {
        int r = idx >> 7, c = idx & 127;
        as[r * 129 + c] = attn[(size_t)(row0 + r) * 128 + c];
    }
    __syncthreads();

    v8f acc = {};
#pragma unroll
    for (int kk = 0; kk < HEADS * DH; kk += 4) {
        v2f a, b;
        a.x = as[lm * 129 + kk + 2 * half];
        a.y = as[lm * 129 + kk + 2 * half + 1];
        const float* bp = w_out + (size_t)(kk + 2 * half) * DIM_X + col0 + lm;
        b.x = bp[0];
        b.y = bp[DIM_X];
        acc = wmma4(a, b, acc);
    }
#pragma unroll
    for (int i = 0; i < 8; ++i) {
        int M = i + 8 * half;
        out[(size_t)(row0 + M) * DIM_X + col0 + lm] = acc[i];
    }
}

// ---------------------------------------------------------------------------
extern "C" void kernel_launch(void* const* d_in, const int* in_sizes, int n_in,
                              void* d_out, int out_size, void* d_ws, size_t ws_size,
                              hipStream_t stream) {
    const float* x     = (const float*)d_in[0];   // [1,2048,512]
    const float* w_q   = (const float*)d_in[1];   // [512,128]
    const float* w_kv  = (const float*)d_in[2];   // [512,256]
    const float* w_out = (const float*)d_in[3];   // [128,512]
    float*       out   = (float*)d_out;           // [1,2048,512]

    float* ws   = (float*)d_ws;
    const size_t per = (size_t)HEADS * N_TOK * DH;   // 262144 floats
    float* qs   = ws;
    float* ks   = ws + per;
    float* vs   = ws + 2 * per;
    float* attn = ws + 3 * per;                       // [2048][128]

    proj_kernel<<<dim3(128, 3), 256, 0, stream>>>(x, w_q, w_kv, qs, ks, vs);
    attn_kernel<<<dim3(HEADS), NW * 32, 0, stream>>>(qs, ks, vs, attn);
    outproj_kernel<<<dim3(128, 4), 256, 0, stream>>>(attn, w_out, out);
}